// Network_86131274154760
// MI455X (gfx1250) — compile-verified
//
#include <hip/hip_runtime.h>
#include <math.h>

typedef __attribute__((ext_vector_type(2))) float v2f;
typedef __attribute__((ext_vector_type(4))) float v4f;
typedef __attribute__((ext_vector_type(8))) float v8f;

// ---------------------------------------------------------------------------
// Kernel 1: zero the 32-float accumulator region of the workspace.
// (harness poisons d_ws with 0xAA; we atomically accumulate into it, so it
//  must be re-zeroed at the start of every launch for replay determinism)
// ---------------------------------------------------------------------------
__global__ void dfm_init_ws(float* __restrict__ ws) {
    if (threadIdx.x < 32) ws[threadIdx.x] = 0.0f;
}

// ---------------------------------------------------------------------------
// Kernel 2: bandwidth-bound streaming reduction over the embedding table.
//   ws[0..15]  += sum_f x[f]   * emb[f, e]
//   ws[16..31] += sum_f x[f]^2 * emb[f, e]^2
// Each thread owns a fixed 4-column chunk (c = tid & 3) of the 16-wide rows,
// so a wave32 loads 8 consecutive rows = 512 contiguous bytes per iteration
// (fully coalesced global_load_b128). emb is single-use -> non-temporal.
// ---------------------------------------------------------------------------
__global__ __launch_bounds__(256) void dfm_reduce(
    const float* __restrict__ x, const float* __restrict__ emb,
    float* __restrict__ ws, int F) {
    __shared__ float s_acc[32];               // [0..15] embf, [16..31] squ
    if (threadIdx.x < 32) s_acc[threadIdx.x] = 0.0f;
    __syncthreads();

    const v4f* emb4 = (const v4f*)emb;        // F*4 float4 chunks
    const int      c       = threadIdx.x & 3; // fixed column chunk 0..3
    const unsigned tid     = blockIdx.x * blockDim.x + threadIdx.x;
    const unsigned row0    = tid >> 2;
    const unsigned rstride = (gridDim.x * blockDim.x) >> 2;

    float a0 = 0.f, a1 = 0.f, a2 = 0.f, a3 = 0.f;  // embf partials
    float q0 = 0.f, q1 = 0.f, q2 = 0.f, q3 = 0.f;  // squ  partials
    for (unsigned r = row0; r < (unsigned)F; r += rstride) {
        float xv = x[r];
        v4f e = __builtin_nontemporal_load(emb4 + ((size_t)r * 4 + c));
        float x2 = xv * xv;
        a0 = fmaf(xv, e.x, a0);
        a1 = fmaf(xv, e.y, a1);
        a2 = fmaf(xv, e.z, a2);
        a3 = fmaf(xv, e.w, a3);
        q0 = fmaf(x2, e.x * e.x, q0);
        q1 = fmaf(x2, e.y * e.y, q1);
        q2 = fmaf(x2, e.z * e.z, q2);
        q3 = fmaf(x2, e.w * e.w, q3);
    }
    // intra-block reduction via LDS float atomics (ds_add_f32)
    atomicAdd(&s_acc[c * 4 + 0], a0);
    atomicAdd(&s_acc[c * 4 + 1], a1);
    atomicAdd(&s_acc[c * 4 + 2], a2);
    atomicAdd(&s_acc[c * 4 + 3], a3);
    atomicAdd(&s_acc[16 + c * 4 + 0], q0);
    atomicAdd(&s_acc[16 + c * 4 + 1], q1);
    atomicAdd(&s_acc[16 + c * 4 + 2], q2);
    atomicAdd(&s_acc[16 + c * 4 + 3], q3);
    __syncthreads();
    // cross-block reduction: 32 global float atomics per block
    if (threadIdx.x < 32) atomicAdd(&ws[threadIdx.x], s_acc[threadIdx.x]);
}

// ---------------------------------------------------------------------------
// Kernel 3: dense head on ONE wave32 using V_WMMA_F32_16X16X4_F32.
//   h1 = relu(W1(32x16) @ embf + b1)    : 2 row tiles x 4 K-steps of 4
//   h2 = relu(W2(32x32) @ h1   + b2)    : 2 row tiles x 8 K-steps of 4
//   out = sigmoid(w_out . [h2, fm, logistic] + b_out)
// B operand broadcasts the activation across all 16 columns, so every
// column of D holds the matvec result. WMMA operand layout (ISA 7.12.2):
//   A 16x4 : lane<16 -> M=lane,    reg j -> K=j   ; lane>=16 -> M=lane-16, K=j+2
//   B 4x16 : reg j holds row K = j + 2*half (value broadcast over columns)
//   D      : reg v holds row M = v + 8*half (identical across the 16 lanes)
// EXEC is all-ones at every WMMA (divergent code only after / reconverged).
// ---------------------------------------------------------------------------
__global__ __launch_bounds__(32) void dfm_head(
    const float* __restrict__ ws,
    const float* __restrict__ w_log, const float* __restrict__ b_log,
    const float* __restrict__ w1, const float* __restrict__ b1,
    const float* __restrict__ w2, const float* __restrict__ b2,
    const float* __restrict__ w_out, const float* __restrict__ b_out,
    float* __restrict__ out) {
    __shared__ float sh1[32];
    __shared__ float sh2[32];
    __shared__ float sfm[16];

    const int lane = threadIdx.x;
    const int half = lane >> 4;   // 0: lanes 0-15, 1: lanes 16-31
    const int m    = lane & 15;   // row within 16-row tile (A operand)
    const int ko   = 2 * half;    // per-lane K offset inside each 4-chunk

    // ---- layer 1: h1 = relu(W1 @ embf + b1), embf = ws[0..15] ----
#pragma unroll
    for (int t = 0; t < 2; ++t) {
        v8f acc = {};
#pragma unroll
        for (int kc = 0; kc < 4; ++kc) {
            const int k = 4 * kc + ko;
            v2f A, B;
            A.x = w1[(16 * t + m) * 16 + k];
            A.y = w1[(16 * t + m) * 16 + k + 1];
            B.x = ws[k];
            B.y = ws[k + 1];
            acc = __builtin_amdgcn_wmma_f32_16x16x4_f32(
                false, A, false, B, (short)0, acc, false, false);
        }
        const int rb = 16 * t + 8 * half;
#pragma unroll
        for (int v = 0; v < 8; ++v) {
            float h = acc[v] + b1[rb + v];
            sh1[rb + v] = h > 0.f ? h : 0.f;  // all lanes write identical value
        }
    }
    __syncthreads();

    // ---- layer 2: h2 = relu(W2 @ h1 + b2) ----
#pragma unroll
    for (int t = 0; t < 2; ++t) {
        v8f acc = {};
#pragma unroll
        for (int kc = 0; kc < 8; ++kc) {
            const int k = 4 * kc + ko;
            v2f A, B;
            A.x = w2[(16 * t + m) * 32 + k];
            A.y = w2[(16 * t + m) * 32 + k + 1];
            B.x = sh1[k];
            B.y = sh1[k + 1];
            acc = __builtin_amdgcn_wmma_f32_16x16x4_f32(
                false, A, false, B, (short)0, acc, false, false);
        }
        const int rb = 16 * t + 8 * half;
#pragma unroll
        for (int v = 0; v < 8; ++v) {
            float h = acc[v] + b2[rb + v];
            sh2[rb + v] = h > 0.f ? h : 0.f;
        }
    }
    __syncthreads();

    // ---- FM second-order term (all WMMAs done; divergence is now fine) ----
    if (lane < 16) {
        float e = ws[lane];
        sfm[lane] = 0.5f * (e * e - ws[16 + lane]);
    }
    __syncthreads();

    // ---- logistic branch + output head + sigmoid (scalar, negligible) ----
    if (lane == 0) {
        float lg = b_log[0];
        for (int e = 0; e < 16; ++e) lg = fmaf(ws[e], w_log[e], lg);
        float s = b_out[0];
        for (int i = 0; i < 32; ++i) s = fmaf(w_out[i], sh2[i], s);
        for (int e = 0; e < 16; ++e) s = fmaf(w_out[32 + e], sfm[e], s);
        s = fmaf(w_out[48], lg, s);
        out[0] = 1.0f / (1.0f + expf(-s));
    }
}

// ---------------------------------------------------------------------------
// Launcher
// inputs (setup_inputs order): x, emb, w_log, b_log, w1, b1, w2, b2, w_out, b_out
// ---------------------------------------------------------------------------
extern "C" void kernel_launch(void* const* d_in, const int* in_sizes, int n_in,
                              void* d_out, int out_size, void* d_ws, size_t ws_size,
                              hipStream_t stream) {
    const float* x     = (const float*)d_in[0];
    const float* emb   = (const float*)d_in[1];
    const float* w_log = (const float*)d_in[2];
    const float* b_log = (const float*)d_in[3];
    const float* w1    = (const float*)d_in[4];
    const float* b1    = (const float*)d_in[5];
    const float* w2    = (const float*)d_in[6];
    const float* b2    = (const float*)d_in[7];
    const float* w_out = (const float*)d_in[8];
    const float* b_out = (const float*)d_in[9];
    float* out = (float*)d_out;
    float* ws  = (float*)d_ws;
    const int F = in_sizes[0];

    dfm_init_ws<<<1, 32, 0, stream>>>(ws);
    // 4096 blocks * 256 threads = 1M threads; 8M float4 loads -> 8 iters/thread.
    dfm_reduce<<<4096, 256, 0, stream>>>(x, emb, ws, F);
    dfm_head<<<1, 32, 0, stream>>>(ws, w_log, b_log, w1, b1, w2, b2,
                                   w_out, b_out, out);
}